// MultiHeadAttention_31714038514411
// MI455X (gfx1250) — compile-verified
//
#include <hip/hip_runtime.h>

// ---------------------------------------------------------------------------
// Low-rank multi-head attention for MI455X (gfx1250), fp32 WMMA throughout.
// D = A(16x4) x B(4x16) + C(16x16) via V_WMMA_F32_16X16X4_F32 (wave32).
// Flash kernel processes 2 query tiles per wave and reads V transposed so
// every B-fragment load is a b64; ~48 WMMA per 24 VMEM in the hot loop.
// ---------------------------------------------------------------------------

typedef float v2f __attribute__((ext_vector_type(2)));
typedef float v8f __attribute__((ext_vector_type(8)));

__device__ __forceinline__ v8f wmma_f32(v2f a, v2f b, v8f c) {
  // 8 args: (neg_a, A, neg_b, B, c_mod, C, reuse_a, reuse_b)
  return __builtin_amdgcn_wmma_f32_16x16x4_f32(
      false, a, false, b, (short)0, c, false, false);
}

#define B_    2
#define S_    1500
#define SP_   1504      // S padded to 16
#define N_    1280
#define H_    20
#define D_    64
#define R_    32
#define BS_   3000      // B*S
#define BSP_  3008      // padded rows
#define ST_   94        // ceil(S/16)
#define PT_   47        // i-tile pairs per head
#define MT_   188       // ceil(BS/16)
#define SCALE_  0.125f  // D^-0.5
#define NEGINF_ (-3.0e38f)

// workspace layout (float offsets)
#define OFF_QKV1 0                                   // [BSP_][96]  q1|k1|v1
#define OFF_WM   (OFF_QKV1 + BSP_*96)                // [H][32][32]
#define OFF_B1   (OFF_WM + H_*32*32)                 // [H][32]
#define OFF_B2   (OFF_B1 + H_*32)                    // [H][32]
#define OFF_B3   (OFF_B2 + H_*32)                    // [H] (padded 32)
#define OFF_VH   (OFF_B3 + 32)                       // [B*H][64][SP_]  (d-major!)
#define OFF_S1   (OFF_VH + B_*H_*SP_*64)             // [B*H][SP_]
#define OFF_D2   (OFF_S1 + B_*H_*SP_)                // [B*H][SP_]
#define OFF_T1   (OFF_D2 + B_*H_*SP_)                // [BSP_][32]

// ---------------------------------------------------------------------------
// Kernel A: per-head merged factors  Wm, b1, b2, b3   (tiny, plain VALU)
// ---------------------------------------------------------------------------
__global__ void precompute_kernel(const float* __restrict__ Wq2,
                                  const float* __restrict__ Wk2,
                                  const float* __restrict__ bq,
                                  const float* __restrict__ bk,
                                  float* __restrict__ ws) {
  const int h = blockIdx.x;
  float* Wm = ws + OFF_WM + h * 1024;
  for (int idx = threadIdx.x; idx < 1024; idx += blockDim.x) {
    const int r = idx >> 5, t = idx & 31;
    float s = 0.f;
    for (int d = 0; d < 64; ++d)
      s += Wq2[(h * 64 + d) * 32 + r] * Wk2[(h * 64 + d) * 32 + t];
    Wm[r * 32 + t] = s;
  }
  if (threadIdx.x < 32) {
    const int r = threadIdx.x;
    float s = 0.f;
    for (int d = 0; d < 64; ++d) s += Wq2[(h * 64 + d) * 32 + r] * bk[h * 64 + d];
    ws[OFF_B1 + h * 32 + r] = s;
  } else if (threadIdx.x < 64) {
    const int t = threadIdx.x - 32;
    float s = 0.f;
    for (int d = 0; d < 64; ++d) s += bq[h * 64 + d] * Wk2[(h * 64 + d) * 32 + t];
    ws[OFF_B2 + h * 32 + t] = s;
  } else if (threadIdx.x == 64) {
    float s = 0.f;
    for (int d = 0; d < 64; ++d) s += bq[h * 64 + d] * bk[h * 64 + d];
    ws[OFF_B3 + h] = s;
  }
}

// ---------------------------------------------------------------------------
// Kernel B: fused qkv rank-32 projections. One wave -> 16 rows x 96 cols.
// out[m, 0:32)=x@Wq1^T, [32:64)=x@Wk1^T, [64:96)=x@Wv1^T
// ---------------------------------------------------------------------------
__global__ void __launch_bounds__(32) qkv_kernel(const float* __restrict__ x,
                                                 const float* __restrict__ Wq1,
                                                 const float* __restrict__ Wk1,
                                                 const float* __restrict__ Wv1,
                                                 float* __restrict__ ws) {
  const int lane = threadIdx.x & 31;
  const int l16 = lane & 15, half = lane >> 4, koff = half * 2;
  const int m0 = blockIdx.x * 16;
  int xr = m0 + l16; if (xr > BS_ - 1) xr = BS_ - 1;   // clamp: keep EXEC full
  const float* xp = x + (long)xr * N_;
  const float* Wtab[6] = { Wq1, Wq1 + 16 * N_, Wk1, Wk1 + 16 * N_, Wv1, Wv1 + 16 * N_ };

  v8f acc[6] = {};
  for (int k0 = 0; k0 < N_; k0 += 4) {
    v2f a; a.x = xp[k0 + koff]; a.y = xp[k0 + koff + 1];
#pragma unroll
    for (int ct = 0; ct < 6; ++ct) {
      const float* wp = Wtab[ct] + (long)l16 * N_ + k0 + koff;  // B[k,n]=W1[n,k]
      v2f bf; bf.x = wp[0]; bf.y = wp[1];
      acc[ct] = wmma_f32(a, bf, acc[ct]);
    }
  }
#pragma unroll
  for (int ct = 0; ct < 6; ++ct)
#pragma unroll
    for (int v = 0; v < 8; ++v)
      ws[OFF_QKV1 + (m0 + v + 8 * half) * 96 + ct * 16 + l16] = acc[ct][v];
}

// ---------------------------------------------------------------------------
// Kernel C: VhT[bh][d][j] = sum_r v1[b,j,r] * Wv2[(h*64+d)*32+r]
// Stored d-major (j contiguous) so flash B-fragment loads are b64.
// ---------------------------------------------------------------------------
__global__ void __launch_bounds__(32) vh_kernel(const float* __restrict__ Wv2,
                                                float* __restrict__ ws) {
  const int lane = threadIdx.x & 31;
  const int l16 = lane & 15, half = lane >> 4, koff = half * 2;
  const int bid = blockIdx.x;
  const int jt = bid % ST_, bh = bid / ST_;
  const int h = bh % H_, b = bh / H_;
  const int j0 = jt * 16;

  const float* v1p = ws + OFF_QKV1 + (b * S_ + j0 + l16) * 96 + 64;  // <= row 3003
  v8f acc[4] = {};
#pragma unroll
  for (int kk = 0; kk < 8; ++kk) {
    const int r = 4 * kk + koff;
    v2f a; a.x = v1p[r]; a.y = v1p[r + 1];
#pragma unroll
    for (int dt = 0; dt < 4; ++dt) {
      const int d = dt * 16 + l16;
      const float* wp = Wv2 + (h * 64 + d) * 32 + r;   // r, r+1 contiguous
      v2f bf; bf.x = wp[0]; bf.y = wp[1];
      acc[dt] = wmma_f32(a, bf, acc[dt]);
    }
  }
  float* VhT = ws + OFF_VH + (long)bh * 64 * SP_;
#pragma unroll
  for (int dt = 0; dt < 4; ++dt) {
    float* vp = VhT + (long)(dt * 16 + l16) * SP_ + j0 + 8 * half;
#pragma unroll
    for (int v = 0; v < 8; ++v)
      vp[v] = acc[dt][v];     // consecutive per lane -> merged wide stores
  }
}

// ---------------------------------------------------------------------------
// Kernel E: per-row bias dots  s1[bh,i]=q1[i,:]·b1[h]   d2[bh,j]=k1[j,:]·b2[h]
// ---------------------------------------------------------------------------
__global__ void bias_kernel(float* __restrict__ ws) {
  const int idx = blockIdx.x * blockDim.x + threadIdx.x;
  if (idx >= B_ * H_ * SP_) return;
  const int s = idx % SP_, bh = idx / SP_;
  const int h = bh % H_, b = bh / H_;
  const float* q1 = ws + OFF_QKV1 + (b * S_ + s) * 96;   // padded rows finite
  const float* b1 = ws + OFF_B1 + h * 32;
  const float* b2 = ws + OFF_B2 + h * 32;
  float s1 = 0.f, d2 = 0.f;
  for (int r = 0; r < 32; ++r) { s1 += q1[r] * b1[r]; d2 += q1[32 + r] * b2[r]; }
  ws[OFF_S1 + idx] = s1;
  ws[OFF_D2 + idx] = d2;
}

// ---------------------------------------------------------------------------
// Kernel D: flash attention, one wave per (b,h, pair-of-i-tiles).
//   qm = Qi @ Wm; stream j-tiles: S = qm @ k1^T (k1 frags shared by both
//   i-tiles), online softmax, O += P @ VhT (V frags shared by both i-tiles).
//   Writes wv into d_out slice [b,i,h*64+d].
// ---------------------------------------------------------------------------
__global__ void __launch_bounds__(32) flash_kernel(float* __restrict__ ws,
                                                   float* __restrict__ wv) {
  __shared__ float ldsQ[2 * 16 * 32];
  __shared__ float ldsP[2 * 16 * 16];
  const int lane = threadIdx.x & 31;
  const int l16 = lane & 15, half = lane >> 4, koff = half * 2;
  const int bid = blockIdx.x;
  const int pt = bid % PT_, bh = bid / PT_;
  const int h = bh % H_, b = bh / H_;
  const int i0 = pt * 32;

  const float* qkv1 = ws + OFF_QKV1;
  const float* Wm   = ws + OFF_WM + h * 1024;
  const float* VhT  = ws + OFF_VH + (long)bh * 64 * SP_;
  const float* s1p  = ws + OFF_S1 + bh * SP_;
  const float* d2p  = ws + OFF_D2 + bh * SP_;
  const float b3h   = ws[OFF_B3 + h];

  // ---- phase 1: qm(u) = Qi(u) @ Wm, reshape C-layout -> A-fragments via LDS
#pragma unroll
  for (int u = 0; u < 2; ++u) {
    const float* qp = qkv1 + (b * S_ + i0 + u * 16 + l16) * 96;  // q1 cols [0,32)
    v2f aQ[8];
#pragma unroll
    for (int kk = 0; kk < 8; ++kk) {
      aQ[kk].x = qp[4 * kk + koff];
      aQ[kk].y = qp[4 * kk + koff + 1];
    }
#pragma unroll
    for (int t2 = 0; t2 < 2; ++t2) {
      v8f acc = {};
      const int t = t2 * 16 + l16;
#pragma unroll
      for (int kk = 0; kk < 8; ++kk) {
        const int r = 4 * kk + koff;
        v2f bf; bf.x = Wm[r * 32 + t]; bf.y = Wm[(r + 1) * 32 + t];
        acc = wmma_f32(aQ[kk], bf, acc);
      }
#pragma unroll
      for (int v = 0; v < 8; ++v)
        ldsQ[u * 512 + (v + 8 * half) * 32 + t] = acc[v];
    }
  }
  __syncthreads();
  v2f aQM[2][8];
#pragma unroll
  for (int u = 0; u < 2; ++u)
#pragma unroll
    for (int kk = 0; kk < 8; ++kk) {
      aQM[u][kk].x = ldsQ[u * 512 + l16 * 32 + 4 * kk + koff];
      aQM[u][kk].y = ldsQ[u * 512 + l16 * 32 + 4 * kk + koff + 1];
    }

  float mrow[2][8], lrow[2][8], s1c[2][8];
#pragma unroll
  for (int u = 0; u < 2; ++u)
#pragma unroll
    for (int v = 0; v < 8; ++v) {
      mrow[u][v] = NEGINF_;
      lrow[u][v] = 0.f;
      s1c[u][v] = (s1p[i0 + u * 16 + v + 8 * half] + b3h) * SCALE_;
    }
  v8f O[2][4] = {};

  // ---- phase 2: stream key/value tiles (shared between the two i-tiles)
  for (int jt = 0; jt < ST_; ++jt) {
    const int j0 = jt * 16;
    // k1 B-fragments, loaded once for both score tiles
    const float* kp = qkv1 + (b * S_ + j0 + l16) * 96 + 32;  // k1 cols [32,64)
    v2f kb[8];
#pragma unroll
    for (int kk = 0; kk < 8; ++kk) {
      kb[kk].x = kp[4 * kk + koff];
      kb[kk].y = kp[4 * kk + koff + 1];
    }
    v8f accS[2] = {};
#pragma unroll
    for (int kk = 0; kk < 8; ++kk) accS[0] = wmma_f32(aQM[0][kk], kb[kk], accS[0]);
#pragma unroll
    for (int kk = 0; kk < 8; ++kk) accS[1] = wmma_f32(aQM[1][kk], kb[kk], accS[1]);

    const int j = j0 + l16;
    const float d2s = d2p[j] * SCALE_;
    const bool jvalid = (j < S_);

#pragma unroll
    for (int u = 0; u < 2; ++u) {
      float pv[8];
#pragma unroll
      for (int v = 0; v < 8; ++v) {
        const float sv = accS[u][v] * SCALE_ + s1c[u][v] + d2s;
        pv[v] = jvalid ? sv : NEGINF_;
      }
      // online softmax per row (row lives in one VGPR across a 16-lane half)
#pragma unroll
      for (int v = 0; v < 8; ++v) {
        float rm = pv[v];
        rm = fmaxf(rm, __shfl_xor(rm, 1, 32));
        rm = fmaxf(rm, __shfl_xor(rm, 2, 32));
        rm = fmaxf(rm, __shfl_xor(rm, 4, 32));
        rm = fmaxf(rm, __shfl_xor(rm, 8, 32));
        const float mnew = fmaxf(mrow[u][v], rm);
        const float alpha = __expf(mrow[u][v] - mnew);
        float p = __expf(pv[v] - mnew);
        float rs = p;
        rs += __shfl_xor(rs, 1, 32);
        rs += __shfl_xor(rs, 2, 32);
        rs += __shfl_xor(rs, 4, 32);
        rs += __shfl_xor(rs, 8, 32);
        lrow[u][v] = lrow[u][v] * alpha + rs;
        mrow[u][v] = mnew;
        pv[v] = p;
#pragma unroll
        for (int dt = 0; dt < 4; ++dt) O[u][dt][v] *= alpha;
      }
#pragma unroll
      for (int v = 0; v < 8; ++v)
        ldsP[u * 256 + (v + 8 * half) * 16 + l16] = pv[v];
    }
    __syncthreads();
    v2f aP[2][4];
#pragma unroll
    for (int u = 0; u < 2; ++u)
#pragma unroll
      for (int kk = 0; kk < 4; ++kk) {
        aP[u][kk].x = ldsP[u * 256 + l16 * 16 + 4 * kk + koff];
        aP[u][kk].y = ldsP[u * 256 + l16 * 16 + 4 * kk + koff + 1];
      }
    __syncthreads();
    // O(u) += P(u) @ VhT tile; V fragments shared by both i-tiles, b64 loads
#pragma unroll
    for (int kk = 0; kk < 4; ++kk) {
      const int jl = j0 + 4 * kk + koff;
#pragma unroll
      for (int dt = 0; dt < 4; ++dt) {
        const float* vp = VhT + (long)(dt * 16 + l16) * SP_ + jl;
        v2f vb; vb.x = vp[0]; vb.y = vp[1];
        O[0][dt] = wmma_f32(aP[0][kk], vb, O[0][dt]);
        O[1][dt] = wmma_f32(aP[1][kk], vb, O[1][dt]);
      }
    }
  }

  // ---- epilogue: normalize and scatter into wv (d_out used as scratch)
#pragma unroll
  for (int u = 0; u < 2; ++u)
#pragma unroll
    for (int v = 0; v < 8; ++v) {
      const int i = i0 + u * 16 + v + 8 * half;
      if (i < S_) {
        const float inv = 1.f / lrow[u][v];
#pragma unroll
        for (int dt = 0; dt < 4; ++dt)
          wv[((long)(b * S_ + i)) * N_ + h * 64 + dt * 16 + l16] = O[u][dt][v] * inv;
      }
    }
}

// ---------------------------------------------------------------------------
// Kernel F1: t1 = (wv + bv) @ Wo1^T    (3000x1280)x(1280x32)
// ---------------------------------------------------------------------------
__global__ void __launch_bounds__(32) oproj1_kernel(const float* __restrict__ wv,
                                                    const float* __restrict__ Wo1,
                                                    const float* __restrict__ bv,
                                                    float* __restrict__ ws) {
  const int lane = threadIdx.x & 31;
  const int l16 = lane & 15, half = lane >> 4, koff = half * 2;
  const int m0 = blockIdx.x * 16;
  int mr = m0 + l16; if (mr > BS_ - 1) mr = BS_ - 1;    // wv has only 3000 rows
  const float* wp = wv + (long)mr * N_;

  v8f acc[2] = {};
  for (int k0 = 0; k0 < N_; k0 += 4) {
    const int k = k0 + koff;
    v2f a; a.x = wp[k] + bv[k]; a.y = wp[k + 1] + bv[k + 1];
#pragma unroll
    for (int nt = 0; nt < 2; ++nt) {
      const float* bp = Wo1 + (long)(nt * 16 + l16) * N_ + k;
      v2f bf; bf.x = bp[0]; bf.y = bp[1];
      acc[nt] = wmma_f32(a, bf, acc[nt]);
    }
  }
  float* t1 = ws + OFF_T1;
#pragma unroll
  for (int nt = 0; nt < 2; ++nt)
#pragma unroll
    for (int v = 0; v < 8; ++v)
      t1[(m0 + v + 8 * half) * 32 + nt * 16 + l16] = acc[nt][v];
}

// ---------------------------------------------------------------------------
// Kernel F2: out = t1 @ Wo2^T + bo     (3000x32)x(32x1280)
// One wave -> 16 rows x 128 cols. grid = 188 * 10
// ---------------------------------------------------------------------------
__global__ void __launch_bounds__(32) oproj2_kernel(const float* __restrict__ ws_c,
                                                    const float* __restrict__ Wo2,
                                                    const float* __restrict__ bo,
                                                    float* __restrict__ out) {
  const int lane = threadIdx.x & 31;
  const int l16 = lane & 15, half = lane >> 4, koff = half * 2;
  const int bid = blockIdx.x;
  const int mt = bid % MT_, g = bid / MT_;
  const int m0 = mt * 16, n0 = g * 128;

  const float* t1 = ws_c + OFF_T1 + (m0 + l16) * 32;   // t1 padded to 3008 rows
  v2f a[8];
#pragma unroll
  for (int kk = 0; kk < 8; ++kk) {
    a[kk].x = t1[4 * kk + koff];
    a[kk].y = t1[4 * kk + koff + 1];
  }
  v8f acc[8] = {};
#pragma unroll
  for (int kk = 0; kk < 8; ++kk) {
    const int r = 4 * kk + koff;
#pragma unroll
    for (int dt = 0; dt < 8; ++dt) {
      const float* bp = Wo2 + (long)(n0 + dt * 16 + l16) * 32 + r;  // B[r,n]=Wo2[n,r]
      v2f bf; bf.x = bp[0]; bf.y = bp[1];
      acc[dt] = wmma_f32(a[kk], bf, acc[dt]);
    }
  }
#pragma unroll
  for (int v = 0; v < 8; ++v) {
    const int m = m0 + v + 8 * half;
    if (m < BS_) {
#pragma unroll
      for (int dt = 0; dt < 8; ++dt) {
        const int n = n0 + dt * 16 + l16;
        out[(long)m * N_ + n] = acc[dt][v] + bo[n];
      }
    }
  }
}

// ---------------------------------------------------------------------------
extern "C" void kernel_launch(void* const* d_in, const int* in_sizes, int n_in,
                              void* d_out, int out_size, void* d_ws, size_t ws_size,
                              hipStream_t stream) {
  (void)in_sizes; (void)n_in; (void)out_size; (void)ws_size;
  const float* x   = (const float*)d_in[0];
  const float* Wq1 = (const float*)d_in[1];
  const float* Wq2 = (const float*)d_in[2];
  const float* bq  = (const float*)d_in[3];
  const float* Wk1 = (const float*)d_in[4];
  const float* Wk2 = (const float*)d_in[5];
  const float* bk  = (const float*)d_in[6];
  const float* Wv1 = (const float*)d_in[7];
  const float* Wv2 = (const float*)d_in[8];
  const float* bv  = (const float*)d_in[9];
  const float* Wo1 = (const float*)d_in[10];
  const float* Wo2 = (const float*)d_in[11];
  const float* bo  = (const float*)d_in[12];
  float* ws  = (float*)d_ws;
  float* out = (float*)d_out;

  precompute_kernel<<<H_, 256, 0, stream>>>(Wq2, Wk2, bq, bk, ws);
  qkv_kernel<<<MT_, 32, 0, stream>>>(x, Wq1, Wk1, Wv1, ws);
  vh_kernel<<<B_ * H_ * ST_, 32, 0, stream>>>(Wv2, ws);
  const int nbias = B_ * H_ * SP_;
  bias_kernel<<<(nbias + 255) / 256, 256, 0, stream>>>(ws);
  flash_kernel<<<B_ * H_ * PT_, 32, 0, stream>>>(ws, out);   // wv -> d_out
  oproj1_kernel<<<MT_, 32, 0, stream>>>(out, Wo1, bv, ws);   // t1 -> ws
  oproj2_kernel<<<MT_ * 10, 32, 0, stream>>>(ws, Wo2, bo, out);
}